// SwitchMoE_10926396801370
// MI455X (gfx1250) — compile-verified
//
#include <hip/hip_runtime.h>
#include <hip/hip_bf16.h>
#include <math.h>

// ---------------- problem constants ----------------
#define D_HID   1024
#define D_FF    4096
#define N_EXP   8
#define N_TOK   8192          // 4 * 2048
#define CAP     1126          // int(8192/8 * 1.1)
#define CAPP    1152          // CAP padded to multiple of 64 (GEMM M tile)

typedef __attribute__((ext_vector_type(2))) float v2f;
typedef __attribute__((ext_vector_type(8))) float v8f;
typedef __attribute__((ext_vector_type(4))) int   v4i;

// ---------------- CDNA5 async global->LDS path (guarded) ----------------
#if defined(__has_builtin)
#if __has_builtin(__builtin_amdgcn_global_load_async_to_lds_b128)
#define MOE_ASYNC_LDS 1
#endif
#endif

#if defined(__has_builtin) && __has_builtin(__builtin_amdgcn_s_wait_asynccnt)
#define MOE_WAIT_ASYNC() __builtin_amdgcn_s_wait_asynccnt(0)
#else
#define MOE_WAIT_ASYNC() asm volatile("s_wait_asynccnt 0x0" ::: "memory")
#endif

// builtin signature (from hipcc diagnostic): (v4i AS1* gsrc, v4i AS3* ldst, imm, imm)
#define MOE_GPTR(p) ((__attribute__((address_space(1))) v4i*)(p))
#define MOE_LPTR(p) ((__attribute__((address_space(3))) v4i*)(p))

// ---------------- workspace layout (bytes) ----------------
static constexpr size_t NBUF_OFF = 0;
static constexpr size_t NBUF_BYTES = (size_t)N_EXP * CAPP * D_HID * 4;     // 37,748,736
static constexpr size_t H_OFF    = NBUF_OFF + NBUF_BYTES;
static constexpr size_t H_BYTES  = (size_t)N_EXP * CAPP * D_FF * 4;        // 150,994,944
static constexpr size_t EIDX_OFF = H_OFF + H_BYTES;
static constexpr size_t POS_OFF  = EIDX_OFF + (size_t)N_TOK * 4;
static constexpr size_t KEEP_OFF = POS_OFF  + (size_t)N_TOK * 4;
static constexpr size_t CNT_OFF  = KEEP_OFF + (size_t)N_TOK * 4;

// ---------------- zero fill ----------------
__global__ void moe_zero_f32(float* __restrict__ p, size_t n4) {
    size_t i = (size_t)blockIdx.x * blockDim.x + threadIdx.x;
    if (i < n4) {
        float4 z = {0.f, 0.f, 0.f, 0.f};
        *(float4*)(p + i * 4) = z;
    }
}

// ---------------- router: logits argmax (softmax is argmax-invariant) ----------------
__global__ void moe_gate(const float* __restrict__ x, const float* __restrict__ gw,
                         int* __restrict__ eidx) {
    __shared__ float gws[D_HID * N_EXP];   // 32 KB
    const int tid = threadIdx.x;
    for (int i = tid; i < D_HID * N_EXP; i += 256) gws[i] = gw[i];
    __syncthreads();

    const int t = blockIdx.x * 256 + tid;
    const float* xr = x + (size_t)t * D_HID;
    float acc[N_EXP];
#pragma unroll
    for (int e = 0; e < N_EXP; ++e) acc[e] = 0.f;
    for (int i = 0; i < D_HID; i += 4) {
        const float4 xv = *(const float4*)(xr + i);
#pragma unroll
        for (int e = 0; e < N_EXP; ++e) {
            acc[e] += xv.x * gws[(i + 0) * N_EXP + e] + xv.y * gws[(i + 1) * N_EXP + e]
                    + xv.z * gws[(i + 2) * N_EXP + e] + xv.w * gws[(i + 3) * N_EXP + e];
        }
    }
    int best = 0; float bv = acc[0];
#pragma unroll
    for (int e = 1; e < N_EXP; ++e) if (acc[e] > bv) { bv = acc[e]; best = e; }
    eidx[t] = best;
}

// ---------------- deterministic ordered scan (single wave32, ballot prefix) ----------------
__global__ void moe_scan(const int* __restrict__ eidx, int* __restrict__ pos,
                         int* __restrict__ keep, float* __restrict__ cnts) {
    const int lane = threadIdx.x;               // 32 lanes, wave32
    int counts[N_EXP];
#pragma unroll
    for (int e = 0; e < N_EXP; ++e) counts[e] = 0;
    const unsigned lt = (1u << lane) - 1u;
    for (int base = 0; base < N_TOK; base += 32) {
        const int t = base + lane;
        const int e = eidx[t];
        int mypos = 0;
#pragma unroll
        for (int ee = 0; ee < N_EXP; ++ee) {
            const unsigned m = (unsigned)__ballot(e == ee);  // uniform across lanes
            if (e == ee) mypos = counts[ee] + __popc(m & lt);
            counts[ee] += __popc(m);
        }
        keep[t] = (mypos < CAP) ? 1 : 0;
        pos[t]  = (mypos < CAP) ? mypos : (CAP - 1);
    }
    if (lane < N_EXP) cnts[lane] = (float)counts[lane];
}

// ---------------- dispatch + LayerNorm scatter ----------------
__global__ void moe_ln_dispatch(const float* __restrict__ x, const int* __restrict__ eidx,
                                const int* __restrict__ pos, const int* __restrict__ keep,
                                const float* __restrict__ ln_g, const float* __restrict__ ln_b,
                                float* __restrict__ nbuf) {
    const int t = blockIdx.x;
    if (!keep[t]) return;                       // row stays zero (never gathered)
    const int e = eidx[t], p = pos[t];
    const int tid = threadIdx.x;                // 256 threads, 4 floats each
    const float4 v = *(const float4*)(x + (size_t)t * D_HID + tid * 4);

    __shared__ float rs[256], rs2[256];
    rs[tid]  = v.x + v.y + v.z + v.w;
    rs2[tid] = v.x * v.x + v.y * v.y + v.z * v.z + v.w * v.w;
    __syncthreads();
    for (int s = 128; s > 0; s >>= 1) {
        if (tid < s) { rs[tid] += rs[tid + s]; rs2[tid] += rs2[tid + s]; }
        __syncthreads();
    }
    const float mu  = rs[0] * (1.f / D_HID);
    const float var = rs2[0] * (1.f / D_HID) - mu * mu;
    const float inv = rsqrtf(var + 1e-5f);

    const float4 g = *(const float4*)(ln_g + (size_t)e * D_HID + tid * 4);
    const float4 b = *(const float4*)(ln_b + (size_t)e * D_HID + tid * 4);
    float4 o;
    o.x = (v.x - mu) * inv * g.x + b.x;
    o.y = (v.y - mu) * inv * g.y + b.y;
    o.z = (v.z - mu) * inv * g.z + b.z;
    o.w = (v.w - mu) * inv * g.w + b.w;
    *(float4*)(nbuf + ((size_t)e * CAPP + p) * D_HID + tid * 4) = o;
}

// ---------------- tiled fp32 WMMA GEMM: C[e] = act(A[e] @ B[e] + bias[e]) ----------------
// A: [E][CAPP][K] row-major, B: [E][K][N] row-major, C: [E][CAPP][N].
// Block: 256 thr (8 waves). Block tile 64M x 128N. K staged 16 at a time, DOUBLE-BUFFERED
// in LDS via GLOBAL_LOAD_ASYNC_TO_LDS_B128 (ASYNCcnt) when available; next slab's HBM
// traffic overlaps current slab's WMMAs. Waves: 4 along M x 2 along N (4 accumulators).
// v_wmma_f32_16x16x4_f32 frags: A (m=lane&15, k=2*(lane>>4)+j), C/D (m=r+8*(lane>>4), n=lane&15).
// LDS strides padded for bank-conflict-free fragment reads (A rows 20 words, B rows 136 words).
#define AS_STR 20
#define BS_STR 136

template <bool APPLY_GELU>
__global__ void moe_ffn_gemm(const float* __restrict__ A, const float* __restrict__ B,
                             const float* __restrict__ bias, float* __restrict__ C,
                             int N, int K) {
    const int e  = blockIdx.z;
    const int m0 = blockIdx.y * 64;
    const int n0 = blockIdx.x * 128;
    const float* Ae = A + (size_t)e * CAPP * K + (size_t)m0 * K;
    const float* Be = B + (size_t)e * K * N + n0;
    float*       Ce = C + (size_t)e * CAPP * N;

    __shared__ float As[2][64 * AS_STR];   // As[buf][m*AS_STR + k]  (k contiguous)
    __shared__ float Bs[2][16 * BS_STR];   // Bs[buf][k*BS_STR + n]  (n contiguous)

    const int tid  = threadIdx.x;
    const int lane = tid & 31;
    const int wid  = tid >> 5;
    const int lo = lane & 15, hi = lane >> 4;
    const int mBase = (wid & 3) * 16;
    const int nBase = (wid >> 2) * 64;

    // staging maps: A -> 1 x b128/thread, B -> 2 x b128/thread (all contiguous copies)
    const int am = tid >> 2;            // 0..63
    const int ak = (tid & 3) * 4;       // 0,4,8,12
    const int bk = (tid * 8) >> 7;      // 0..15
    const int bn = (tid * 8) & 127;     // 0..120 step 8

    v8f acc[4];
#pragma unroll
    for (int i = 0; i < 4; ++i) acc[i] = (v8f){};

    auto stage = [&](int buf, int k0) {
        const float* ga  = Ae + (size_t)am * K + k0 + ak;
        const float* gb  = Be + (size_t)(k0 + bk) * N + bn;
        float* la  = &As[buf][am * AS_STR + ak];
        float* lb0 = &Bs[buf][bk * BS_STR + bn];
#if defined(MOE_ASYNC_LDS)
        __builtin_amdgcn_global_load_async_to_lds_b128(MOE_GPTR(ga),     MOE_LPTR(la),      0, 0);
        __builtin_amdgcn_global_load_async_to_lds_b128(MOE_GPTR(gb),     MOE_LPTR(lb0),     0, 0);
        __builtin_amdgcn_global_load_async_to_lds_b128(MOE_GPTR(gb + 4), MOE_LPTR(lb0 + 4), 0, 0);
#else
        const float4 av  = *(const float4*)ga;
        const float4 bv0 = *(const float4*)gb;
        const float4 bv1 = *(const float4*)(gb + 4);
        *(float4*)la        = av;
        *(float4*)lb0       = bv0;
        *(float4*)(lb0 + 4) = bv1;
#endif
    };

    stage(0, 0);
#if defined(MOE_ASYNC_LDS)
    MOE_WAIT_ASYNC();
#endif
    __syncthreads();

    for (int k0 = 0; k0 < K; k0 += 16) {
        const int  cur  = (k0 >> 4) & 1;
        const bool more = (k0 + 16) < K;
        if (more) stage(cur ^ 1, k0 + 16);     // overlap next slab copy with this slab's math

        const float* Ac = &As[cur][0];
        const float* Bc = &Bs[cur][0];
#pragma unroll
        for (int kk = 0; kk < 16; kk += 4) {
            const v2f a = *(const v2f*)(Ac + (mBase + lo) * AS_STR + kk + 2 * hi);
#pragma unroll
            for (int ns = 0; ns < 4; ++ns) {
                v2f b;
                b[0] = Bc[(kk + 2 * hi + 0) * BS_STR + nBase + ns * 16 + lo];
                b[1] = Bc[(kk + 2 * hi + 1) * BS_STR + nBase + ns * 16 + lo];
                acc[ns] = __builtin_amdgcn_wmma_f32_16x16x4_f32(
                    false, a, false, b, (short)0, acc[ns], false, false);
            }
        }
        if (more) {
#if defined(MOE_ASYNC_LDS)
            MOE_WAIT_ASYNC();                   // next slab fully landed in LDS
#endif
            __syncthreads();                    // everyone done reading cur, next staged
        }
    }

    // epilogue: bias + (optional) exact GELU
#pragma unroll
    for (int ns = 0; ns < 4; ++ns) {
        const int n = n0 + nBase + ns * 16 + lo;
        const float bs = bias[(size_t)e * N + n];
#pragma unroll
        for (int r = 0; r < 8; ++r) {
            const int m = m0 + mBase + r + 8 * hi;
            float v = acc[ns][r] + bs;
            if (APPLY_GELU) v = 0.5f * v * (1.f + erff(v * 0.70710678118654752f));
            Ce[(size_t)m * N + n] = v;
        }
    }
}

// ---------------- gather back to token order ----------------
__global__ void moe_gather(const float* __restrict__ y, const int* __restrict__ eidx,
                           const int* __restrict__ pos, const int* __restrict__ keep,
                           float* __restrict__ out) {
    const int t = blockIdx.x;
    const int tid = threadIdx.x;
    float4 r = {0.f, 0.f, 0.f, 0.f};
    if (keep[t]) {
        r = *(const float4*)(y + ((size_t)eidx[t] * CAPP + pos[t]) * D_HID + tid * 4);
    }
    *(float4*)(out + (size_t)t * D_HID + tid * 4) = r;
}

// ---------------- load-balance loss ----------------
__global__ void moe_loss(const float* __restrict__ cnts, float* __restrict__ loss_out) {
    if (threadIdx.x != 0 || blockIdx.x != 0) return;
    const float u = (float)N_TOK / (float)N_EXP;   // 1024
    float dot = 0.f, nc = 0.f;
#pragma unroll
    for (int e = 0; e < N_EXP; ++e) { dot += cnts[e] * u; nc += cnts[e] * cnts[e]; }
    const float nu = u * sqrtf((float)N_EXP);
    const float cosv = dot / (sqrtf(nc) * nu + 1e-8f);
    loss_out[0] = 1.f - cosv * 0.01f;
}

// ---------------- host: launch sequence ----------------
extern "C" void kernel_launch(void* const* d_in, const int* in_sizes, int n_in,
                              void* d_out, int out_size, void* d_ws, size_t ws_size,
                              hipStream_t stream) {
    (void)in_sizes; (void)n_in; (void)out_size; (void)ws_size;
    const float* x      = (const float*)d_in[0];
    const float* gate_w = (const float*)d_in[1];
    const float* ln_g   = (const float*)d_in[2];
    const float* ln_b   = (const float*)d_in[3];
    const float* w1     = (const float*)d_in[4];
    const float* b1     = (const float*)d_in[5];
    const float* w2     = (const float*)d_in[6];
    const float* b2     = (const float*)d_in[7];
    float* out = (float*)d_out;                        // [8192*1024] output + [1] loss

    char* ws = (char*)d_ws;
    float* nbuf = (float*)(ws + NBUF_OFF);             // also reused as y after GEMM2
    float* hbuf = (float*)(ws + H_OFF);
    int*   eidx = (int*)(ws + EIDX_OFF);
    int*   pos  = (int*)(ws + POS_OFF);
    int*   keep = (int*)(ws + KEEP_OFF);
    float* cnts = (float*)(ws + CNT_OFF);

    // 1) zero dispatch buffer (unfilled capacity rows must be zero)
    {
        const size_t n4 = (size_t)N_EXP * CAPP * D_HID / 4;
        moe_zero_f32<<<(unsigned)((n4 + 255) / 256), 256, 0, stream>>>(nbuf, n4);
    }
    // 2) router argmax
    moe_gate<<<N_TOK / 256, 256, 0, stream>>>(x, gate_w, eidx);
    // 3) deterministic ordered capacity scan (single wave32)
    moe_scan<<<1, 32, 0, stream>>>(eidx, pos, keep, cnts);
    // 4) LayerNorm + scatter dispatch
    moe_ln_dispatch<<<N_TOK, 256, 0, stream>>>(x, eidx, pos, keep, ln_g, ln_b, nbuf);
    // 5) FFN GEMM1 + GELU: h = gelu(nb @ w1 + b1)   [E][CAPP][D_FF]
    {
        dim3 grid(D_FF / 128, CAPP / 64, N_EXP);
        moe_ffn_gemm<true><<<grid, 256, 0, stream>>>(nbuf, w1, b1, hbuf, D_FF, D_HID);
    }
    // 6) FFN GEMM2: y = h @ w2 + b2   [E][CAPP][D_HID]  (y reuses nbuf region)
    {
        dim3 grid(D_HID / 128, CAPP / 64, N_EXP);
        moe_ffn_gemm<false><<<grid, 256, 0, stream>>>(hbuf, w2, b2, nbuf, D_HID, D_FF);
    }
    // 7) gather (dropped tokens -> zeros)
    moe_gather<<<N_TOK, 256, 0, stream>>>(nbuf, eidx, pos, keep, out);
    // 8) scalar load-balancing loss
    moe_loss<<<1, 32, 0, stream>>>(cnts, out + (size_t)N_TOK * D_HID);
}